// BisectionLoss_53120155517189
// MI455X (gfx1250) — compile-verified
//
#include <hip/hip_runtime.h>
#include <cfloat>
#include <cstdint>

// ---------------------------------------------------------------------------
// entmax-1.5 bisection loss, one workgroup per row.
//   B = 4096 rows, V = 32000 cols, 50 bisection iterations.
// Row (128 KB) is staged once into LDS via the CDNA5 Tensor Data Mover, then
// 32 elements/thread live in VGPRs for the whole transcendental-bound loop.
// ---------------------------------------------------------------------------

#define TPB   1024
#define VDIM  32000
#define VPAD  32768           // 32 elems/thread * 1024 threads
#define NITER 50
#define INV_EXP (1.0f / 31999.0f)     // reference uses 1/(d-1) as written
#define DIFF0   0.99440983f           // 1 - 32000^(1-1.5) = 1 - 32000^-0.5

typedef unsigned int u32x4 __attribute__((ext_vector_type(4)));
typedef int          i32x8 __attribute__((ext_vector_type(8)));
typedef int          i32x4 __attribute__((ext_vector_type(4)));

__device__ __forceinline__ float pow_inv(float d) {
  // relu(d)^(1/31999) = exp2(log2(max(d,0)) * INV_EXP)
  // d<=0: v_log_f32(0) = -inf -> exp2(-inf) = 0  (branchless clamp)
  float z = fmaxf(d, 0.0f);
  return __builtin_amdgcn_exp2f(INV_EXP * __builtin_amdgcn_logf(z));
}

__device__ __forceinline__ float wave_sum(float v) {
#pragma unroll
  for (int m = 16; m >= 1; m >>= 1) v += __shfl_xor(v, m, 32);
  return v;
}
__device__ __forceinline__ float wave_max(float v) {
#pragma unroll
  for (int m = 16; m >= 1; m >>= 1) v = fmaxf(v, __shfl_xor(v, m, 32));
  return v;
}

// Fixed-order (deterministic) sum of one 32-float reduction row via 8x b128.
__device__ __forceinline__ float row_sum32(const float* r) {
  const float4* r4 = (const float4*)r;
  float s = 0.0f;
#pragma unroll
  for (int j = 0; j < 8; ++j) {
    float4 v = r4[j];
    s += ((v.x + v.y) + (v.z + v.w));
  }
  return s;
}

__global__ __launch_bounds__(TPB, 1) void entmax_bisect_loss_kernel(
    const float* __restrict__ X, const int* __restrict__ target,
    float* __restrict__ out) {
  __shared__ __align__(16) float xrow[VPAD];   // 131072 B
  __shared__ __align__(16) float red[4][32];   // reduction scratch

  const int row  = blockIdx.x;
  const int tid  = threadIdx.x;
  const int lane = tid & 31;
  const int wav  = tid >> 5;

  // Thread 0 fetches X[row, target[row]] early (hides latency behind loop).
  float x_tgt = 0.0f;
  if (tid == 0) {
    int tg = target[row];
    x_tgt = X[(size_t)row * VDIM + (size_t)tg];
  }

#if __has_builtin(__builtin_amdgcn_tensor_load_to_lds)
  if (tid < 32) {  // wave 0 issues one TDM descriptor: 1 x 32000 f32 tile
    uint64_t ga      = (uint64_t)(uintptr_t)X + (uint64_t)row * (VDIM * 4ull);
    uint32_t lds_off = (uint32_t)(uintptr_t)&xrow[0];   // low 32b of LDS flat addr
    u32x4 g0 = { 1u,                                    // count = 1 (valid D#)
                 lds_off,                               // lds_addr
                 (uint32_t)ga,                          // global_addr[31:0]
                 ((uint32_t)(ga >> 32) & 0x01FFFFFFu)   // global_addr[56:32]
                   | 0x80000000u };                     // type = 2 ("image")
    i32x8 g1 = { 0x00020000,   // data_size = 2 (4 bytes)
                 0x7D000000,   // tensor_dim0 = 32000 (low16 in bits 31:16)
                 0x00010000,   // tensor_dim0 hi16 = 0 ; tensor_dim1 = 1
                 0x7D000000,   // tile_dim0 = 32000
                 0x00000001,   // tile_dim1 = 1 ; tile_dim2 = 0
                 0x00007D00,   // tensor_dim0_stride = 32000
                 0, 0 };
    i32x4 g2 = {0, 0, 0, 0};   // groups 2/3 unused (2D tensor)
    i32x4 g3 = {0, 0, 0, 0};
    i32x8 g4 = {0, 0, 0, 0, 0, 0, 0, 0};
    __builtin_amdgcn_tensor_load_to_lds(g0, g1, g2, g3, g4, 0);
    __builtin_amdgcn_s_wait_tensorcnt(0);
  }
#else
  {
    const float4* __restrict__ src = (const float4*)(X + (size_t)row * VDIM);
    float4* dst = (float4*)xrow;
    for (int i = tid; i < VDIM / 4; i += TPB) dst[i] = src[i];
  }
#endif
  // Pad tail [32000, 32768) with a large-negative sentinel (contributes 0).
  if (tid < VPAD - VDIM) xrow[VDIM + tid] = -FLT_MAX;
  __syncthreads();

  // 32 elements per thread into VGPRs; Xs = (alpha-1)*X = 0.5*X
  float xs[32];
  float m = -FLT_MAX;
  const float4* xr4 = (const float4*)xrow;
#pragma unroll
  for (int k = 0; k < 8; ++k) {
    float4 v = xr4[tid + TPB * k];
    xs[4 * k + 0] = 0.5f * v.x;
    xs[4 * k + 1] = 0.5f * v.y;
    xs[4 * k + 2] = 0.5f * v.z;
    xs[4 * k + 3] = 0.5f * v.w;
    m = fmaxf(m, fmaxf(fmaxf(xs[4 * k], xs[4 * k + 1]),
                       fmaxf(xs[4 * k + 2], xs[4 * k + 3])));
  }

  // Workgroup max (deterministic: fixed-order combine of 32 partials)
  m = wave_max(m);
  if (lane == 0) red[0][wav] = m;
  __syncthreads();
  {
    const float4* r4 = (const float4*)&red[0][0];
    m = -FLT_MAX;
#pragma unroll
    for (int j = 0; j < 8; ++j) {
      float4 v = r4[j];
      m = fmaxf(m, fmaxf(fmaxf(v.x, v.y), fmaxf(v.z, v.w)));
    }
  }

  // Bisection: double-buffered reduction scratch -> one barrier per iteration
  float t_min = m - 1.0f;
  float diff  = DIFF0;
  float t = 0.0f, S = 0.0f;
  for (int i = 0; i < NITER; ++i) {
    diff *= 0.5f;
    t = t_min + diff;
    // 4 independent accumulators: adds overlap the quarter-rate v_log/v_exp
    float s0 = 0.0f, s1 = 0.0f, s2 = 0.0f, s3 = 0.0f;
#pragma unroll
    for (int k = 0; k < 8; ++k) {
      s0 += pow_inv(xs[4 * k + 0] - t);
      s1 += pow_inv(xs[4 * k + 1] - t);
      s2 += pow_inv(xs[4 * k + 2] - t);
      s3 += pow_inv(xs[4 * k + 3] - t);
    }
    float s = wave_sum((s0 + s1) + (s2 + s3));
    const int buf = 1 + (i & 1);
    if (lane == 0) red[buf][wav] = s;
    __syncthreads();
    S = row_sum32(&red[buf][0]);
    if (S - 1.0f >= 0.0f) t_min = t;
  }

  // p = Z/S from the LAST iteration's t and S (matches reference exactly)
  const float invS = __builtin_amdgcn_rcpf(S);
  float a1 = 0.0f, a2 = 0.0f;
#pragma unroll
  for (int k = 0; k < 32; ++k) {
    float z = pow_inv(xs[k] - t);
    float p = z * invS;
    a1 += p * __builtin_amdgcn_sqrtf(p);   // p^1.5
    a2 += p * (2.0f * xs[k]);              // p * X   (X = 2*Xs)
  }
  a1 = wave_sum(a1);
  a2 = wave_sum(a2);
  if (lane == 0) { red[0][wav] = a1; red[3][wav] = a2; }
  __syncthreads();
  if (tid == 0) {
    float T1 = row_sum32(&red[0][0]);
    float T2 = row_sum32(&red[3][0]);
    // loss = (1 - sum p^1.5)/(1.5*0.5) + (p·X - X[target])
    out[row] = (1.0f - T1) * (1.0f / 0.75f) + (T2 - x_tgt);
  }
}

extern "C" void kernel_launch(void* const* d_in, const int* in_sizes, int n_in,
                              void* d_out, int out_size, void* d_ws,
                              size_t ws_size, hipStream_t stream) {
  (void)n_in; (void)out_size; (void)d_ws; (void)ws_size;
  const float* X      = (const float*)d_in[0];
  const int*   target = (const int*)d_in[1];
  float*       out    = (float*)d_out;
  const int B = in_sizes[0] / VDIM;   // 4096
  entmax_bisect_loss_kernel<<<dim3(B), dim3(TPB), 0, stream>>>(X, target, out);
}